// NystromLinearKernel_57501022158911
// MI455X (gfx1250) — compile-verified
//
#include <hip/hip_runtime.h>

// Problem constants from the reference
#define BB    8
#define CIN   32
#define COUT  32
#define HH    128
#define WW    128
#define SS    1024
#define JJ    (HH * WW)

typedef __attribute__((ext_vector_type(2))) float v2f;
typedef __attribute__((ext_vector_type(8))) float v8f;

// ---------------------------------------------------------------------------
// Stage 1: R[b, m, i] = sum_s coef(m, s) * v[b, i, idx[s]],  m in {0:1, 1:x_s, 2:y_s}
// Formulated as a 16x16x4 f32 WMMA GEMM: A = coef matrix (16 x K, rows>=3 zero),
// B = gathered samples (K x 16 channels). One wave per (batch, channel-tile).
// A layout (32b, 16x4): lane l(0..15)/l+16 holds M=l; VGPR0 -> K=0/2, VGPR1 -> K=1/3.
// B layout (4x16): VGPR0 -> K=0 (lanes0-15) / K=2 (lanes16-31), N = lane&15.
// ---------------------------------------------------------------------------
__global__ __launch_bounds__(32) void nystrom_stats_wmma(
    const float* __restrict__ v, const int* __restrict__ idx,
    float* __restrict__ R) {
  const int b     = blockIdx.x >> 1;
  const int n0    = (blockIdx.x & 1) * 16;
  const int lane  = threadIdx.x;
  const int half  = lane >> 4;   // 0: K pair {0,1}; 1: K pair {2,3}
  const int l     = lane & 15;
  const float sc  = 2.0f / 127.0f;

  const float* __restrict__ vch = v + ((size_t)b * CIN + (n0 + l)) * (size_t)JJ;

  v8f c = {};
  for (int k0 = 0; k0 < SS; k0 += 4) {
    const int kA = k0 + 2 * half;
    // two consecutive sample indices for this lane's K pair
    const int2 s2 = *(const int2*)(idx + kA);
    const int s0 = s2.x, s1 = s2.y;

    const float x0 = (float)(s0 & 127) * sc - 1.0f;
    const float y0 = (float)(s0 >> 7)  * sc - 1.0f;
    const float x1 = (float)(s1 & 127) * sc - 1.0f;
    const float y1 = (float)(s1 >> 7)  * sc - 1.0f;

    v2f a;  // coef(M=l, K=kA), coef(M=l, K=kA+1); rows 3..15 are zero padding
    a.x = (l == 0) ? 1.0f : (l == 1) ? x0 : (l == 2) ? y0 : 0.0f;
    a.y = (l == 0) ? 1.0f : (l == 1) ? x1 : (l == 2) ? y1 : 0.0f;

    v2f bm; // gathered sample values for channel n0+l at K=kA, kA+1
    bm.x = vch[s0];
    bm.y = vch[s1];

    // D = A x B + C  (f32 16x16x4)
    c = __builtin_amdgcn_wmma_f32_16x16x4_f32(
        /*neg_a=*/false, a, /*neg_b=*/false, bm,
        /*c_mod=*/(short)0, c, /*reuse_a=*/false, /*reuse_b=*/false);
  }

  // C/D layout: VGPR j, lanes 0-15 -> M=j, N=l.  Rows 0..2 hold R0,R1,R2.
  if (lane < 16) {
    float* __restrict__ Rb = R + b * (3 * CIN);
    Rb[0 * CIN + n0 + l] = c[0];
    Rb[1 * CIN + n0 + l] = c[1];
    Rb[2 * CIN + n0 + l] = c[2];
  }
}

// ---------------------------------------------------------------------------
// Stage 2: per (b,o) affine coefficients from R and weight (COUT, CIN, 2).
// ---------------------------------------------------------------------------
__global__ __launch_bounds__(256) void nystrom_coeffs(
    const float* __restrict__ R, const float* __restrict__ w,
    float* __restrict__ C) {
  const int t = blockIdx.x * blockDim.x + threadIdx.x;
  if (t >= BB * COUT) return;
  const int b = t >> 5;
  const int o = t & 31;
  const float* __restrict__ Rb = R + b * (3 * CIN);
  const float* __restrict__ wo = w + o * (CIN * 2);
  float cx = 0.0f, cy = 0.0f, t0 = 0.0f;
  #pragma unroll
  for (int i = 0; i < CIN; ++i) {
    const float w0 = wo[i * 2 + 0];
    const float w1 = wo[i * 2 + 1];
    cx = fmaf(Rb[0 * CIN + i], w0, cx);
    cy = fmaf(Rb[0 * CIN + i], w1, cy);
    t0 = fmaf(Rb[1 * CIN + i], w0, t0);
    t0 = fmaf(Rb[2 * CIN + i], w1, t0);
  }
  C[t * 3 + 0] = cx;
  C[t * 3 + 1] = cy;
  C[t * 3 + 2] = -t0;   // C0 = -T
}

// ---------------------------------------------------------------------------
// Stage 3: streaming fill  u[bo, y, x] = x*cx + y*cy + c0  with b128 stores.
// One block per (b,o) plane; 256 threads x 16 float4 iters = 16384 pixels.
// ---------------------------------------------------------------------------
__global__ __launch_bounds__(256) void nystrom_fill(
    const float* __restrict__ C, float* __restrict__ out) {
  const int bo = blockIdx.x;
  const float cx = C[bo * 3 + 0];
  const float cy = C[bo * 3 + 1];
  const float c0 = C[bo * 3 + 2];
  float* __restrict__ po = out + (size_t)bo * JJ;
  const float sc = 2.0f / 127.0f;

  for (int i = threadIdx.x; i < JJ / 4; i += 256) {
    const int r   = i * 4;              // 4 consecutive pixels, same row
    const int row = r >> 7;
    const int col = r & 127;
    const float y  = (float)row * sc - 1.0f;
    const float xb = (float)col * sc - 1.0f;
    const float base = fmaf(cy, y, c0);
    float4 val;
    val.x = fmaf(cx, xb,             base);
    val.y = fmaf(cx, xb + sc,        base);
    val.z = fmaf(cx, xb + 2.0f * sc, base);
    val.w = fmaf(cx, xb + 3.0f * sc, base);
    ((float4*)po)[i] = val;
  }
}

// ---------------------------------------------------------------------------
extern "C" void kernel_launch(void* const* d_in, const int* in_sizes, int n_in,
                              void* d_out, int out_size, void* d_ws, size_t ws_size,
                              hipStream_t stream) {
  (void)in_sizes; (void)n_in; (void)out_size; (void)ws_size;
  const float* v   = (const float*)d_in[0];   // (B, CIN, H, W) f32
  const float* w   = (const float*)d_in[1];   // (COUT, CIN, 2) f32
  const int*   idx = (const int*)d_in[2];     // (S,) int
  float*       out = (float*)d_out;           // (B, COUT, H, W) f32

  float* R    = (float*)d_ws;                       // B*3*CIN floats
  float* Cofs = R + BB * 3 * CIN;                   // B*COUT*3 floats

  nystrom_stats_wmma<<<BB * 2, 32, 0, stream>>>(v, idx, R);
  nystrom_coeffs<<<1, 256, 0, stream>>>(R, w, Cofs);
  nystrom_fill<<<BB * COUT, 256, 0, stream>>>(Cofs, out);
}